// LAMAFace_11201274708636
// MI455X (gfx1250) — compile-verified
//
#include <hip/hip_runtime.h>
#include <hip/hip_bf16.h>
#include <math.h>

// ---------------------------------------------------------------------------
// LAMAFace feature-norm normalization for MI455X (gfx1250, wave32).
//
// Key insight: the reference's kernel_norm ([512 x 70722] = 145 MB read) and
// all proxy_mb updates are dead code w.r.t. the returned tensor. Only ~70 KB
// of data actually feeds the output, so this is a latency-bound micro-kernel.
//
// Ring-buffer semantics: all B pushes happen before any normalization, so each
// sample's final bank row is reconstructed independently from the original
// feature_mb row + in-order overwrites at slots (counts[lb]+k) % Q.
//
// The per-row mean / unbiased-variance reductions (Q=30 padded to 32) are done
// with V_WMMA_F32_16X16X4_F32 row-sums (A = 16x4 f32 tile, B = all-ones),
// accumulating f32-exactly over 8 K-chunks. Two passes: sum -> mean, then
// masked (x-mean)^2 -> variance.
// ---------------------------------------------------------------------------

typedef __attribute__((ext_vector_type(2))) float v2f;
typedef __attribute__((ext_vector_type(8))) float v8f;

#define B_TOTAL            512
#define QSZ                30
#define QPAD               32
#define SAMPLES_PER_BLOCK  64
#define THREADS_PER_BLOCK  128     // 4 waves; each wave owns a 16-sample WMMA tile
#define ROWSTRIDE          33      // pad to avoid LDS bank conflicts
#define EPS                1e-3f

__global__ __launch_bounds__(THREADS_PER_BLOCK)
void lamaface_norm_kernel(const float* __restrict__ feature_norm,  // [B]
                          const int*   __restrict__ label,         // [B]
                          const float* __restrict__ feature_mb,    // [C,Q]
                          const int*   __restrict__ counts,        // [C]
                          float*       __restrict__ out)           // [B]
{
    __shared__ int   s_label[B_TOTAL];
    __shared__ float s_fn[B_TOTAL];
    __shared__ float s_bank[SAMPLES_PER_BLOCK][ROWSTRIDE];
    __shared__ float s_red[SAMPLES_PER_BLOCK];   // wmma row-sum results
    __shared__ float s_mean[SAMPLES_PER_BLOCK];
    __shared__ int   s_n[SAMPLES_PER_BLOCK];

    const int tid  = threadIdx.x;
    const int base = blockIdx.x * SAMPLES_PER_BLOCK;

    // ---- Stage all labels + feature norms (pushes from ANY sample can hit
    //      this block's rows, so every block needs the full batch). ----------
    for (int i = tid; i < B_TOTAL; i += THREADS_PER_BLOCK) {
        s_label[i] = label[i];
        s_fn[i]    = feature_norm[i];
    }
    __syncthreads();

    // ---- Reconstruct each sample's FINAL bank row (post all B pushes). ----
    if (tid < SAMPLES_PER_BLOCK) {
        const int s  = base + tid;
        const int lb = s_label[s];
        const int c0 = counts[lb];
        const float* row = feature_mb + (long long)lb * QSZ;
        #pragma unroll
        for (int q = 0; q < QSZ; ++q) s_bank[tid][q] = row[q];
        s_bank[tid][30] = 0.0f;
        s_bank[tid][31] = 0.0f;

        int k = 0;
        for (int i = 0; i < B_TOTAL; ++i) {          // in-order, last write wins
            if (s_label[i] == lb) {
                s_bank[tid][(c0 + k) % QSZ] = s_fn[i];
                ++k;
            }
        }
        int n = c0 + k;
        if (n > QSZ) n = QSZ;                        // n >= 1 (own push)
        s_n[tid] = n;
        for (int q = n; q < QSZ; ++q) s_bank[tid][q] = 0.0f;  // mask for sum
    }
    __syncthreads();

    // ---- WMMA tile geometry: wave w reduces local samples [w*16, w*16+16).
    //      A-matrix 16x4 f32 layout: lane = (K/2)*16 + M, VGPR = K%2. --------
    const int wave  = tid >> 5;
    const int lane  = tid & 31;
    const int m     = lane & 15;
    const int khalf = lane >> 4;              // which K-pair this lane holds
    const int sm    = wave * 16 + m;          // local sample row for this lane

    v2f ones; ones.x = 1.0f; ones.y = 1.0f;   // B = all-ones 4x16 => row sums

    // ---- Pass 1: row sums -> mean. EXEC all-1s through the WMMA loop. -----
    {
        v8f acc = {};
        #pragma unroll
        for (int kc = 0; kc < 8; ++kc) {
            const int k0 = kc * 4 + khalf * 2;
            v2f a;
            a.x = s_bank[sm][k0];
            a.y = s_bank[sm][k0 + 1];
            acc = __builtin_amdgcn_wmma_f32_16x16x4_f32(
                false, a, false, ones, (short)0, acc, false, false);
        }
        // D layout: VGPR r = M=r (lanes 0-15) / M=8+r (lanes 16-31); value is
        // identical across N since B is all-ones, so lanes 0 and 16 suffice.
        if (lane == 0) {
            #pragma unroll
            for (int r = 0; r < 8; ++r) s_red[wave * 16 + r] = acc[r];
        }
        if (lane == 16) {
            #pragma unroll
            for (int r = 0; r < 8; ++r) s_red[wave * 16 + 8 + r] = acc[r];
        }
    }
    __syncthreads();

    // ---- mean, then stage masked centered-squares for the variance pass. --
    if (tid < SAMPLES_PER_BLOCK) {
        const int n = s_n[tid];
        const float mean = s_red[tid] / (float)n;
        s_mean[tid] = mean;
        #pragma unroll
        for (int q = 0; q < QPAD; ++q) {
            float v = (q < n) ? (s_bank[tid][q] - mean) : 0.0f;
            s_bank[tid][q] = v * v;
        }
    }
    __syncthreads();

    // ---- Pass 2: row sums of squared deviations -> unbiased variance. -----
    {
        v8f acc = {};
        #pragma unroll
        for (int kc = 0; kc < 8; ++kc) {
            const int k0 = kc * 4 + khalf * 2;
            v2f a;
            a.x = s_bank[sm][k0];
            a.y = s_bank[sm][k0 + 1];
            acc = __builtin_amdgcn_wmma_f32_16x16x4_f32(
                false, a, false, ones, (short)0, acc, false, false);
        }
        if (lane == 0) {
            #pragma unroll
            for (int r = 0; r < 8; ++r) s_red[wave * 16 + r] = acc[r];
        }
        if (lane == 16) {
            #pragma unroll
            for (int r = 0; r < 8; ++r) s_red[wave * 16 + 8 + r] = acc[r];
        }
    }
    __syncthreads();

    // ---- Final normalization + store. -------------------------------------
    if (tid < SAMPLES_PER_BLOCK) {
        const int s = base + tid;
        const int n = s_n[tid];
        const float denom_n = fmaxf((float)n - 1.0f, 1.0f);   // torch unbiased std
        const float stdv    = sqrtf(s_red[tid] / denom_n);
        const float centered = s_fn[s] - s_mean[tid];
        const float res = (n > 2) ? (centered / (stdv + EPS))
                                  : (centered * 0.05f);        // centered / 20
        out[s] = res;
    }
}

extern "C" void kernel_launch(void* const* d_in, const int* in_sizes, int n_in,
                              void* d_out, int out_size, void* d_ws, size_t ws_size,
                              hipStream_t stream) {
    // setup_inputs() order:
    //   0: feature_norm [512,1] f32
    //   1: label        [512]   int
    //   2: kernel       [512,70722] f32   (dead w.r.t. output -> never read)
    //   3: feature_mb   [70722,30] f32
    //   4: proxy_mb     [70722,30] f32    (dead w.r.t. output -> never read)
    //   5: counts       [70722] int32
    const float* feature_norm = (const float*)d_in[0];
    const int*   label        = (const int*)d_in[1];
    const float* feature_mb   = (const float*)d_in[3];
    const int*   counts       = (const int*)d_in[5];
    float* out = (float*)d_out;

    const int blocks = B_TOTAL / SAMPLES_PER_BLOCK;   // 8
    lamaface_norm_kernel<<<blocks, THREADS_PER_BLOCK, 0, stream>>>(
        feature_norm, label, feature_mb, counts, out);
}